// Linear_self_63333587747446
// MI455X (gfx1250) — compile-verified
//
#include <hip/hip_runtime.h>

typedef __attribute__((ext_vector_type(2))) float    v2f;
typedef __attribute__((ext_vector_type(8))) float    v8f;
typedef __attribute__((ext_vector_type(4))) unsigned u32x4;
typedef __attribute__((ext_vector_type(4))) int      i32x4;
typedef __attribute__((ext_vector_type(8))) int      i32x8;

#define Mdim 512   // OUT
#define Kdim 512   // F
#define Tdim 256   // T
#define BM 128
#define BN 64
#define BK 32
#define LDW 36     // W tile [m][k]: 32 dwords + 4 pad  (TDM pad_interval=4, pad_amount=3)
#define LDB 68     // x tile [k][t]: 64 dwords + 4 pad  (TDM pad_interval=5, pad_amount=3)
#define KTILES (Kdim / BK)   // 16

__device__ __forceinline__ unsigned lds_off(const void* p) {
    // low 32 bits of the generic address == LDS byte address
    return (unsigned)(unsigned long long)p;
}

#if __has_builtin(__builtin_amdgcn_tensor_load_to_lds)
#define USE_TDM 1
#else
#define USE_TDM 0
#endif

#if USE_TDM
// One TDM descriptor moves a whole 2D tile global->LDS, with LDS row padding.
__device__ __forceinline__ void tdm_load_2d(unsigned lds_addr, const void* gaddr,
                                            unsigned td0, unsigned td1,
                                            unsigned tile_d0, unsigned tile_d1,
                                            unsigned stride0,
                                            unsigned pad_interval, unsigned pad_amount)
{
    const unsigned long long ga = (unsigned long long)gaddr;
    u32x4 g0;
    g0.x = 1u;                                    // count=1 (valid), user mode
    g0.y = lds_addr;                              // D#.lds_addr
    g0.z = (unsigned)ga;                          // global_addr[31:0]
    g0.w = (unsigned)((ga >> 32) & 0x01FFFFFFull) // global_addr[56:32]
         | (2u << 30);                            // type = 2 ("image")

    i32x8 g1;
    g1[0] = (int)((2u << 16)                      // data_size = 4 bytes
                | (1u << 20)                      // pad_enable
                | (pad_interval << 22)            // dwords-per-row before pad (log2-1)
                | (pad_amount << 25));            // pad dwords - 1
    g1[1] = (int)((td0 & 0xFFFFu) << 16);                          // tensor_dim0[15:0]
    g1[2] = (int)(((td0 >> 16) & 0xFFFFu) | ((td1 & 0xFFFFu) << 16));
    g1[3] = (int)(((td1 >> 16) & 0xFFFFu) | ((tile_d0 & 0xFFFFu) << 16)); // tile_dim0
    g1[4] = (int)(tile_d1 & 0xFFFFu);                              // tile_dim1 (tile_dim2=0)
    g1[5] = (int)stride0;                                          // tensor_dim0_stride[31:0]
    g1[6] = 0;
    g1[7] = 0;

    const i32x4 gz = {0, 0, 0, 0};                // groups 2/3 unused for 2D tiles
#if __clang_major__ >= 23
    const i32x8 gz8 = {0, 0, 0, 0, 0, 0, 0, 0};
    __builtin_amdgcn_tensor_load_to_lds(g0, g1, gz, gz, gz8, 0);
#else
    __builtin_amdgcn_tensor_load_to_lds(g0, g1, gz, gz, 0);
#endif
}
#else
// Fallback: per-thread async copies (ASYNCcnt-tracked).
__device__ __forceinline__ void async_b128(unsigned dst_lds, const float* src) {
    asm volatile("global_load_async_to_lds_b128 %0, %1, off"
                 :: "v"(dst_lds), "v"(src) : "memory");
}
__device__ __forceinline__ void wait_async0() {
    asm volatile("s_wait_asynccnt 0x0" ::: "memory");
}
#endif

__device__ __forceinline__ void stage_tile(const float* __restrict__ W,
                                           const float* __restrict__ Xp,
                                           int tileM, int tileT, int k0,
                                           float* sWb, float* sXb, int tid, int wave) {
#if USE_TDM
    if (wave == 0) {
        // W tile: rows m (tile_d1=128), row length 32 dwords (tile_d0), stride 512
        tdm_load_2d(lds_off(sWb), W + (long)tileM * Kdim + k0,
                    Kdim, Mdim, BK, BM, Kdim, /*pad_interval=*/4, /*pad_amount=*/3);
        // x tile: rows k (tile_d1=32), row length 64 dwords (tile_d0), stride 256
        tdm_load_2d(lds_off(sXb), Xp + (long)k0 * Tdim + tileT,
                    Tdim, Kdim, BN, BK, Tdim, /*pad_interval=*/5, /*pad_amount=*/3);
    }
#else
    #pragma unroll
    for (int i = 0; i < 4; ++i) {
        int idx = tid + i * 256;
        int m   = idx >> 3;
        int k4  = (idx & 7) * 4;
        async_b128(lds_off(&sWb[m * LDW + k4]),
                   W + (long)(tileM + m) * Kdim + k0 + k4);
    }
    #pragma unroll
    for (int i = 0; i < 2; ++i) {
        int idx = tid + i * 256;
        int kk  = idx >> 4;
        int t4  = (idx & 15) * 4;
        async_b128(lds_off(&sXb[kk * LDB + t4]),
                   Xp + (long)(k0 + kk) * Tdim + tileT + t4);
    }
#endif
}

__device__ __forceinline__ void stage_wait(int wave) {
#if USE_TDM
    if (wave == 0) __builtin_amdgcn_s_wait_tensorcnt((short)0);
#else
    wait_async0();
#endif
}

__global__ __launch_bounds__(256) void linear_wmma_f32(
    const float* __restrict__ x, const float* __restrict__ W,
    const float* __restrict__ bias, float* __restrict__ out)
{
    __shared__ __align__(16) float sW[2][BM * LDW];   // double-buffered [m][k]
    __shared__ __align__(16) float sX[2][BK * LDB];   // double-buffered [k][t]

    const int tid  = threadIdx.x;
    const int lane = tid & 31;
    const int wave = tid >> 5;
    const int lm   = lane & 15;   // M/N index within 16-wide fragment
    const int lh   = lane >> 4;   // lane half: K = 2*lh + vgpr

    const int wm = (wave & 3) * 32;   // wave M offset inside block tile
    const int wn = (wave >> 2) * 32;  // wave T offset inside block tile

    const int tileM = (blockIdx.x & 3) * BM;
    const int tileT = (blockIdx.x >> 2) * BN;
    const long bn   = blockIdx.y;     // flattened (B,N)

    const float* Xp = x   + bn * (long)(Kdim * Tdim);
    float*       Cp = out + bn * (long)(Mdim * Tdim);

    v8f acc[2][2] = {};

    // prologue: stage tile 0 into buffer 0
    stage_tile(W, Xp, tileM, tileT, 0, sW[0], sX[0], tid, wave);
    stage_wait(wave);
    __syncthreads();

    for (int kt = 0; kt < KTILES; ++kt) {
        const float* sWc = sW[kt & 1];
        const float* sXc = sX[kt & 1];

        // prefetch next K-tile into the other buffer while we compute
        if (kt + 1 < KTILES)
            stage_tile(W, Xp, tileM, tileT, (kt + 1) * BK,
                       sW[(kt + 1) & 1], sX[(kt + 1) & 1], tid, wave);

        #pragma unroll
        for (int kk = 0; kk < BK; kk += 4) {
            v2f a[2], b[2];
            #pragma unroll
            for (int mt = 0; mt < 2; ++mt)
                a[mt] = *(const v2f*)(&sWc[(wm + mt * 16 + lm) * LDW + kk + 2 * lh]);
            #pragma unroll
            for (int nt = 0; nt < 2; ++nt) {
                const int t = wn + nt * 16 + lm;
                v2f bf;
                bf.x = sXc[(kk + 2 * lh) * LDB + t];
                bf.y = sXc[(kk + 2 * lh + 1) * LDB + t];
                b[nt] = bf;
            }
            #pragma unroll
            for (int mt = 0; mt < 2; ++mt)
                #pragma unroll
                for (int nt = 0; nt < 2; ++nt)
                    acc[mt][nt] = __builtin_amdgcn_wmma_f32_16x16x4_f32(
                        false, a[mt], false, b[nt],
                        (short)0, acc[mt][nt], false, false);
        }

        stage_wait(wave);   // next tile's DMA has landed in LDS
        __syncthreads();    // all waves done reading current buffer
    }

    // epilogue: bias add + non-temporal store (out is write-once, keep L2 for x)
    #pragma unroll
    for (int mt = 0; mt < 2; ++mt) {
        #pragma unroll
        for (int r = 0; r < 8; ++r) {
            const int m  = tileM + wm + mt * 16 + r + 8 * lh;
            const float bv = bias[m];
            #pragma unroll
            for (int nt = 0; nt < 2; ++nt) {
                const int tcol = tileT + wn + nt * 16 + lm;
                __builtin_nontemporal_store(acc[mt][nt][r] + bv,
                                            &Cp[(long)m * Tdim + tcol]);
            }
        }
    }
}

extern "C" void kernel_launch(void* const* d_in, const int* in_sizes, int n_in,
                              void* d_out, int out_size, void* d_ws, size_t ws_size,
                              hipStream_t stream) {
    const float* x  = (const float*)d_in[0];  // (4,256,512,256) f32
    const float* W  = (const float*)d_in[1];  // (512,512) f32
    const float* b  = (const float*)d_in[2];  // (512,) f32
    float* out      = (float*)d_out;          // (4,256,512,256) f32

    dim3 grid(16, 1024);   // 16 output tiles per (b,n) x 1024 (b,n) pairs
    linear_wmma_f32<<<grid, 256, 0, stream>>>(x, W, b, out);
}